// LF_Layer_53833120088137
// MI455X (gfx1250) — compile-verified
//
#include <hip/hip_runtime.h>
#include <hip/hip_bf16.h>

typedef __attribute__((ext_vector_type(2))) float v2f;
typedef __attribute__((ext_vector_type(8))) float v8f;

// Fast hardware reciprocal: single v_rcp_f32 (TRANS), ~1 ulp.
__device__ __forceinline__ float frcp(float x) { return __builtin_amdgcn_rcpf(x); }

// Accumulate sum_{i in quarter q} 1/(D[i] + D[i+1]) for one row.
// Quarter q covers interface terms i = 256q .. 256q+255 (q<3) or 768..1022 (q=3).
__device__ __forceinline__ float quarter_hsum(const float* __restrict__ row, int q) {
    const int c0 = q << 8;                       // 256*q
    const float4* __restrict__ vp = reinterpret_cast<const float4*>(row + c0);

    // gfx1250 prefetch path: pull the next quarter's head while we stream this one
    if (q < 3) __builtin_prefetch(row + c0 + 256, 0, 1);

    float4 v = vp[0];
    float acc = frcp(v.x + v.y);
    acc      += frcp(v.y + v.z);
    acc      += frcp(v.z + v.w);
    float prevw = v.w;

    #pragma unroll 8
    for (int j = 1; j < 64; ++j) {
        v = vp[j];
        acc += frcp(prevw + v.x);
        acc += frcp(v.x + v.y);
        acc += frcp(v.y + v.z);
        acc += frcp(v.z + v.w);
        prevw = v.w;
    }
    // closing term pairs last element of this quarter with first of the next
    if (q < 3) acc += frcp(prevw + row[c0 + 256]);
    return acc;
}

// One wave32 handles a 16-row tile. Per-lane partial harmonic sums are arranged
// exactly as the V_WMMA_F32_16X16X4_F32 A-matrix (16 rows x 4 K-quarters):
//   lanes 0-15 : M = lane,    VGPR0 = K0 (cols 0..256),  VGPR1 = K1 (256..512)
//   lanes 16-31: M = lane-16, VGPR0 = K2 (512..768),     VGPR1 = K3 (768..1023)
// A x ones(4x16) reduces the K dimension for all 16 rows in one matrix op.
__global__ __launch_bounds__(256) void peds_harmonic_wmma(
        const float* __restrict__ D, float* __restrict__ K, int nrows) {
    const int lane  = threadIdx.x & 31;
    const int wave  = threadIdx.x >> 5;
    const int tile0 = (blockIdx.x * (blockDim.x >> 5) + wave) << 4;  // first row of tile
    if (tile0 >= nrows) return;  // wave-uniform guard (EXEC stays all-1s)

    const int m  = lane & 15;    // row within tile
    const int hi = lane >> 4;    // 0 -> quarters 0,1 ; 1 -> quarters 2,3
    const float* __restrict__ row = D + (size_t)(tile0 + m) * 1024;

    const float p0 = quarter_hsum(row, 2 * hi);
    const float p1 = quarter_hsum(row, 2 * hi + 1);

    v2f a; a[0] = p0;   a[1] = p1;
    v2f b; b[0] = 1.0f; b[1] = 1.0f;     // ones: D[m][n] = sum_k A[m][k]
    v8f c = {};
    c = __builtin_amdgcn_wmma_f32_16x16x4_f32(
            /*neg_a=*/false, a, /*neg_b=*/false, b,
            /*c_mod=*/(short)0, c, /*reuse_a=*/false, /*reuse_b=*/false);

    // C/D layout: VGPR r, lanes 0-15 -> M=r (N=lane); lanes 16-31 -> M=8+r.
    // Column N=0 of each half suffices: lane 0 writes rows 0..7, lane 16 rows 8..15.
    // K = (N-1)/Σ 1/Dh = 1023/(2*Σ 1/(Di+Di+1)) = 511.5/sum.
    if (m == 0) {
        float* __restrict__ o = K + tile0 + (hi << 3);
        #pragma unroll
        for (int r = 0; r < 8; ++r) o[r] = 511.5f * frcp(c[r]);
    }
}

extern "C" void kernel_launch(void* const* d_in, const int* in_sizes, int n_in,
                              void* d_out, int out_size, void* d_ws, size_t ws_size,
                              hipStream_t stream) {
    const float* D = (const float*)d_in[0];
    float* K = (float*)d_out;
    const int nrows = out_size;              // 16384 rows of 1024 floats
    const int waves_per_block = 8;           // 256 threads
    const int rows_per_block  = waves_per_block * 16;
    const int grid = (nrows + rows_per_block - 1) / rows_per_block;
    peds_harmonic_wmma<<<grid, waves_per_block * 32, 0, stream>>>(D, K, nrows);
}